// ContrastByClassCalculator_89730456748384
// MI455X (gfx1250) — compile-verified
//
#include <hip/hip_runtime.h>
#include <hip/hip_bf16.h>

// MoCo-style per-class contrastive loss, fused for MI455X (gfx1250).
// Memory-bound: queue = 100 MB f32 streams once (grouped by class);
// l_neg tiles via V_WMMA_F32_16X16X4_F32; log-softmax done online per lane
// (no cross-lane traffic in the hot loop), merged once at the end.
// ~100-130 MB traffic -> ~5us at 23.3 TB/s.

#define C_NUM   100
#define NSAMP   512
#define DDIM    128
#define KDIM    2048
// sum_c ceil(n_c/16) <= floor(512/16) + 100 = 132; round up.
#define MAX_TILES 160

typedef float v2f __attribute__((ext_vector_type(2)));
typedef float v8f __attribute__((ext_vector_type(8)));

// ---------------------------------------------------------------------------
// Kernel 1: bucket samples by class and cut into 16-row WMMA tiles.
// ---------------------------------------------------------------------------
__global__ __launch_bounds__(256) void build_tiles_kernel(
    const int* __restrict__ labels,
    int* __restrict__ ntiles_p,
    int* __restrict__ tile_class,
    int* __restrict__ tile_idx,
    int* __restrict__ order)
{
    __shared__ int cnt[C_NUM];
    __shared__ int offs[C_NUM];
    __shared__ int cur[C_NUM];
    __shared__ int tb[C_NUM];
    const int tid = threadIdx.x;

    for (int c = tid; c < C_NUM; c += 256) cnt[c] = 0;
    __syncthreads();
    for (int n = tid; n < NSAMP; n += 256) atomicAdd(&cnt[labels[n]], 1);
    __syncthreads();
    if (tid == 0) {
        int off = 0, toff = 0;
        for (int c = 0; c < C_NUM; ++c) {
            offs[c] = off; cur[c] = off; off += cnt[c];
            tb[c] = toff; toff += (cnt[c] + 15) >> 4;
        }
        *ntiles_p = toff;
    }
    __syncthreads();
    // scatter sample ids per class (intra-class order is arbitrary; per-row
    // results are independent of tile membership, so output is deterministic)
    for (int n = tid; n < NSAMP; n += 256) {
        int pos = atomicAdd(&cur[labels[n]], 1);
        order[pos] = n;
    }
    __syncthreads();
    for (int c = tid; c < C_NUM; c += 256) {
        const int n_c = cnt[c];
        const int nt  = (n_c + 15) >> 4;
        for (int t = 0; t < nt; ++t) {
            const int gt = tb[c] + t;
            tile_class[gt] = c;
            for (int j = 0; j < 16; ++j) {
                const int li = t * 16 + j;
                tile_idx[gt * 16 + j] = (li < n_c) ? order[offs[c] + li] : -1;
            }
        }
    }
}

// ---------------------------------------------------------------------------
// Kernel 2: fused WMMA GEMM + online log-sum-exp.
// One block (8 wave32) per sample-tile. Wave w owns k-tiles {w, w+8, ...}.
// A/B per-lane layouts per CDNA5 ISA 7.12.2 (32-bit 16x4 A, 4x16 B).
// A fragments are hoisted into registers once per wave (invariant over kt);
// softmax stats kept per-lane in the loop, cross-lane merged once at the end.
// ---------------------------------------------------------------------------
__global__ __launch_bounds__(256) void moco_fused_kernel(
    const float* __restrict__ q,
    const float* __restrict__ k,
    const float* __restrict__ queue,
    const int*  __restrict__ ntiles_p,
    const int*  __restrict__ tile_class,
    const int*  __restrict__ tile_idx,
    float* __restrict__ loss_out)
{
    const float invT = 1.0f / 0.07f;
    const int t = blockIdx.x;
    if (t >= *ntiles_p) return;

    __shared__ int   s_idx[16];
    __shared__ int   s_cls;
    __shared__ float s_lpos[16];
    __shared__ float s_m[8][16];
    __shared__ float s_s[8][16];

    const int tid = threadIdx.x;
    if (tid < 16)  s_idx[tid] = tile_idx[t * 16 + tid];
    if (tid == 16) s_cls = tile_class[t];
    __syncthreads();

    // l_pos for the (up to) 16 samples of this tile: threads 0..15.
    if (tid < 16) {
        const int s = s_idx[tid];
        float acc = 0.0f;
        if (s >= 0) {
            const float* qp = q + s * DDIM;
            const float* kp = k + s * DDIM;
            #pragma unroll 8
            for (int d = 0; d < DDIM; d += 4) {
                const float4 qa = *(const float4*)(qp + d);
                const float4 ka = *(const float4*)(kp + d);
                acc += qa.x * ka.x + qa.y * ka.y + qa.z * ka.z + qa.w * ka.w;
            }
        }
        s_lpos[tid] = acc;
    }

    const int wave   = tid >> 5;
    const int lane   = tid & 31;
    const int mrowid = lane & 15;      // A row M / B,D column N for this lane
    const int hi     = lane >> 4;      // lane group: K base = 2*hi

    const int   samp  = s_idx[mrowid];
    const float amask = (samp >= 0) ? 1.0f : 0.0f;   // zero padded rows, keep EXEC full
    const float* qrow  = q + (size_t)((samp >= 0) ? samp : 0) * DDIM + 2 * hi;
    const float* Bbase = queue + (size_t)s_cls * DDIM * KDIM;

    // Preload this tile's A fragments once: 32 x v2f = 64 VGPRs, reused for
    // all 16 ktiles. Inner loop becomes {2 queue loads + 1 WMMA} only.
    v2f areg[DDIM / 4];
    #pragma unroll
    for (int d4 = 0; d4 < DDIM / 4; ++d4) {
        areg[d4].x = qrow[d4 * 4]     * amask;
        areg[d4].y = qrow[d4 * 4 + 1] * amask;
    }

    // Per-lane running stats: lane holds (m, s) for row r (+8 if hi) at its
    // own column; columns are merged once after the loop.
    float m_l[8], s_l[8];
    #pragma unroll
    for (int r = 0; r < 8; ++r) { m_l[r] = -__builtin_inff(); s_l[r] = 0.0f; }

    for (int kt = wave; kt < KDIM / 16; kt += 8) {
        const int kk0 = kt * 16;
        const float* bcol = Bbase + (size_t)2 * hi * KDIM + kk0 + mrowid;

        v8f acc = {0.f, 0.f, 0.f, 0.f, 0.f, 0.f, 0.f, 0.f};
        #pragma unroll
        for (int d4 = 0; d4 < DDIM / 4; ++d4) {
            v2f b;                       // B: queue[c, 4*d4+2*hi(+1), kk0+n]
            b.x = bcol[(size_t)(d4 * 4)     * KDIM];
            b.y = bcol[(size_t)(d4 * 4 + 1) * KDIM];
            acc = __builtin_amdgcn_wmma_f32_16x16x4_f32(
                false, areg[d4], false, b, (short)0, acc, false, false);
        }

        // Pure-VALU online update (no cross-lane traffic in the hot loop).
        #pragma unroll
        for (int r = 0; r < 8; ++r) {
            const float v  = acc[r] * invT;
            const float mn = fmaxf(m_l[r], v);
            s_l[r] = s_l[r] * __expf(m_l[r] - mn) + __expf(v - mn);
            m_l[r] = mn;
        }
    }

    // Merge (m,s) pairs across the 16 columns of each half-wave: 4 butterfly
    // steps, done once per wave (was: 8 butterflies per ktile).
    #pragma unroll
    for (int r = 0; r < 8; ++r) {
        float m = m_l[r], s = s_l[r];
        #pragma unroll
        for (int mask = 8; mask >= 1; mask >>= 1) {
            const float om = __shfl_xor(m, mask, 16);
            const float os = __shfl_xor(s, mask, 16);
            const float mn = fmaxf(m, om);
            s = s * __expf(m - mn) + os * __expf(om - mn);
            m = mn;
        }
        m_l[r] = m; s_l[r] = s;
    }

    // Stats are replicated across each 16-lane group; one writer per group.
    if ((lane & 15) == 0) {
        #pragma unroll
        for (int r = 0; r < 8; ++r) {
            s_m[wave][r + 8 * hi] = m_l[r];
            s_s[wave][r + 8 * hi] = s_l[r];
        }
    }
    __syncthreads();

    // Combine 8 waves + the l_pos logit; emit per-sample loss.
    if (tid < 16) {
        const int s = s_idx[tid];
        if (s >= 0) {
            float m = -__builtin_inff();
            #pragma unroll
            for (int w = 0; w < 8; ++w) m = fmaxf(m, s_m[w][tid]);
            float ssum = 0.0f;
            #pragma unroll
            for (int w = 0; w < 8; ++w) ssum += s_s[w][tid] * __expf(s_m[w][tid] - m);
            const float lp  = s_lpos[tid] * invT;
            const float m2  = fmaxf(m, lp);
            const float lse = m2 + __logf(ssum * __expf(m - m2) + __expf(lp - m2));
            loss_out[s] = lse - lp;
        }
    }
}

// ---------------------------------------------------------------------------
// Kernel 3: deterministic mean over 512 per-sample losses.
// ---------------------------------------------------------------------------
__global__ __launch_bounds__(256) void reduce_mean_kernel(
    const float* __restrict__ loss, float* __restrict__ out)
{
    __shared__ float sh[256];
    const int tid = threadIdx.x;
    sh[tid] = loss[tid] + loss[tid + 256];
    __syncthreads();
    for (int s = 128; s > 0; s >>= 1) {
        if (tid < s) sh[tid] += sh[tid + s];
        __syncthreads();
    }
    if (tid == 0) out[0] = sh[0] * (1.0f / (float)NSAMP);
}

// ---------------------------------------------------------------------------
extern "C" void kernel_launch(void* const* d_in, const int* in_sizes, int n_in,
                              void* d_out, int out_size, void* d_ws, size_t ws_size,
                              hipStream_t stream) {
    const float* q      = (const float*)d_in[0];   // (512,128)
    const float* k      = (const float*)d_in[1];   // (512,128)
    const float* queue  = (const float*)d_in[2];   // (100,128,2048)
    // d_in[3] = class_weights: unused by the reference computation
    const int*   labels = (const int*)d_in[4];     // (512,)

    char* ws = (char*)d_ws;
    int*   ntiles     = (int*)(ws + 0);
    int*   tile_class = (int*)(ws + 64);
    int*   tile_idx   = (int*)(ws + 768);                       // MAX_TILES*16 ints
    int*   order      = (int*)(ws + 768 + MAX_TILES * 16 * 4);  // 512 ints
    float* loss       = (float*)(ws + 768 + MAX_TILES * 16 * 4 + NSAMP * 4);

    build_tiles_kernel<<<1, 256, 0, stream>>>(labels, ntiles, tile_class, tile_idx, order);
    moco_fused_kernel<<<MAX_TILES, 256, 0, stream>>>(q, k, queue, ntiles, tile_class,
                                                     tile_idx, loss);
    reduce_mean_kernel<<<1, 256, 0, stream>>>(loss, (float*)d_out);
}